// PEAttention_58171037057236
// MI455X (gfx1250) — compile-verified
//
#include <hip/hip_runtime.h>
#include <hip/hip_bf16.h>

// out = xf @ Wv.T  (the reference's attention branch reduces to a factor of 1.0;
// see analysis). Pure fp32 GEMM, M=8192, K=256, N=256, done with
// V_WMMA_F32_16X16X4_F32 so precision matches the fp32 reference.

typedef __attribute__((ext_vector_type(2))) float v2f;
typedef __attribute__((ext_vector_type(4))) float v4f;
typedef __attribute__((ext_vector_type(8))) float v8f;

#define M_TOTAL 8192   // B*N*QD*DB = 2*128*8*4
#define KDIM    256    // C (contraction)
#define NOUT    256    // C (output)
#define KC      64     // K chunk staged in LDS
#define LDSTR   68     // padded LDS row stride (floats): conflict-free frag reads
#define NROWS_B 128    // Wv rows (output cols) per block

__global__ __launch_bounds__(256)
void PEAttention_v_gemm_f32wmma(const float* __restrict__ x,
                                const float* __restrict__ Wv,
                                float* __restrict__ out) {
  __shared__ float As[16 * LDSTR];        //  4.35 KB
  __shared__ float Bs[NROWS_B * LDSTR];   // 34.8  KB

  const int t    = threadIdx.x;
  const int lane = t & 31;
  const int wave = t >> 5;        // 0..7
  const int l    = lane & 15;     // row within fragment
  const int half = lane >> 4;     // 0: K=0,1  1: K=2,3 (fp32 16x4 A layout)

  const int m0   = blockIdx.x * 16;        // output row tile
  const int nb   = blockIdx.y * NROWS_B;   // output col group base
  const int nLoc = wave * 16;              // this wave's 16-col subtile

  v8f acc = {0.f, 0.f, 0.f, 0.f, 0.f, 0.f, 0.f, 0.f};

  for (int k0 = 0; k0 < KDIM; k0 += KC) {
    // ---- stage A tile (16 x KC), coalesced float4: 256 threads x 1 ----
    {
      const int row = t >> 4;              // 0..15
      const int c4  = (t & 15) << 2;       // 0..60
      v4f va = *(const v4f*)(x + (size_t)(m0 + row) * KDIM + k0 + c4);
      *(v4f*)(&As[row * LDSTR + c4]) = va;
    }
    // ---- stage B tile (Wv rows nb..nb+127, cols k0..k0+63): 8 float4/thread ----
    #pragma unroll
    for (int i = 0; i < 8; ++i) {
      const int j   = t + i * 256;         // 0..2047
      const int row = j >> 4;              // 0..127
      const int c4  = (j & 15) << 2;
      v4f vb = *(const v4f*)(Wv + (size_t)(nb + row) * KDIM + k0 + c4);
      *(v4f*)(&Bs[row * LDSTR + c4]) = vb;
    }
    __syncthreads();

    // ---- 16 fp32 WMMA steps over this K chunk ----
    // A frag: lane(l,half) holds A[m0+l, kk+2*half .. +1]
    // B frag: lane(l,half) holds Wv[nb+nLoc+l, kk+2*half .. +1]
    const float* aRow = &As[l * LDSTR + 2 * half];
    const float* bRow = &Bs[(nLoc + l) * LDSTR + 2 * half];
    #pragma unroll
    for (int kc = 0; kc < KC; kc += 4) {
      v2f af = *(const v2f*)(aRow + kc);
      v2f bf = *(const v2f*)(bRow + kc);
      // 8 args: (neg_a, A, neg_b, B, c_mod, C, reuse_a, reuse_b)
      acc = __builtin_amdgcn_wmma_f32_16x16x4_f32(false, af, false, bf,
                                                  (short)0, acc, false, false);
    }
    __syncthreads();
  }

  // ---- store 16x16 f32 C/D tile: VGPR i -> row (m0 + i + 8*half), col (nb+nLoc+l)
  const int col = nb + nLoc + l;
  #pragma unroll
  for (int i = 0; i < 8; ++i) {
    out[(size_t)(m0 + i + 8 * half) * NOUT + col] = acc[i];
  }
}

extern "C" void kernel_launch(void* const* d_in, const int* in_sizes, int n_in,
                              void* d_out, int out_size, void* d_ws, size_t ws_size,
                              hipStream_t stream) {
  // setup_inputs order: x, mask, Wq, Wk, Wv. Only x and Wv are needed
  // (attention branch contributes exactly a factor of 1.0 per analysis).
  (void)in_sizes; (void)n_in; (void)out_size; (void)d_ws; (void)ws_size;
  const float* x  = (const float*)d_in[0];
  const float* Wv = (const float*)d_in[4];
  float* out = (float*)d_out;

  dim3 grid(M_TOTAL / 16, NOUT / NROWS_B, 1);  // 512 x 2 blocks
  dim3 block(256, 1, 1);                       // 8 wave32s
  PEAttention_v_gemm_f32wmma<<<grid, block, 0, stream>>>(x, Wv, out);
}